// STKFLoss_2559800509143
// MI455X (gfx1250) — compile-verified
//
#include <hip/hip_runtime.h>

typedef float v2f __attribute__((ext_vector_type(2)));
typedef float v4f __attribute__((ext_vector_type(4)));
typedef float v8f __attribute__((ext_vector_type(8)));

#define BETA_C   (1.0f / 9.0f)
#define INV_BETA 9.0f
#define EPS_C    1e-6f

// Fast reciprocal: v_rcp_f32 + one Newton-Raphson step (~0.5 ulp), branchless.
__device__ __forceinline__ float rcp_nr(float x) {
    float r = __builtin_amdgcn_rcpf(x);
    r = r * __builtin_fmaf(-x, r, 2.0f);
    return r;
}

__device__ __forceinline__ void sigma_v(float w, float h, float r,
                                        float& s11, float& s12, float& s22) {
    w = fminf(fmaxf(w, 1e-7f), 1e7f);
    h = fminf(fmaxf(h, 1e-7f), 1e7f);
    float c, s;
    __sincosf(r, &s, &c);
    float a = 0.25f * w * w;   // (0.5*w)^2
    float b = 0.25f * h * h;   // (0.5*h)^2
    s11 = a * c * c + b * s * s;
    s22 = a * s * s + b * c * c;
    s12 = (a - b) * s * c;
}

__device__ __forceinline__ float loss_from(const float p[5], const float t[5],
                                           const float sp[5], const float st[5]) {
    // Smooth-L1 on xy: 0.5*d*d/BETA == 4.5*d*d (multiply, no divide, no branch)
    float dx = fabsf(p[0] - t[0]);
    float dy = fabsf(p[1] - t[1]);
    float lx = (dx < BETA_C) ? (0.5f * INV_BETA) * dx * dx : dx - 0.5f * BETA_C;
    float ly = (dy < BETA_C) ? (0.5f * INV_BETA) * dy * dy : dy - 0.5f * BETA_C;
    float xy_loss = lx + ly;

    float p11, p12, p22, t11, t12, t22;
    sigma_v(sp[2], sp[3], sp[4], p11, p12, p22);
    sigma_v(st[2], st[3], st[4], t11, t12, t22);

    float Vb_p = 4.0f * sqrtf(p11 * p22 - p12 * p12);
    float Vb_t = 4.0f * sqrtf(t11 * t22 - t12 * t12);

    float u11 = p11 + t11, u12 = p12 + t12, u22 = p22 + t22;
    float det_u = u11 * u22 - u12 * u12;
    float inv = rcp_nr(det_u);
    float i11 =  u22 * inv;
    float i12 = -u12 * inv;
    float i22 =  u11 * inv;

    float k11 = p11 * i11 + p12 * i12;
    float k12 = p11 * i12 + p12 * i22;
    float k21 = p12 * i11 + p22 * i12;
    float k22 = p12 * i12 + p22 * i22;

    float s11 = p11 - (k11 * p11 + k12 * p12);
    float s12 = p12 - (k11 * p12 + k12 * p22);
    float s21 = p12 - (k21 * p11 + k22 * p12);
    float s22 = p22 - (k21 * p12 + k22 * p22);

    float Vb = 4.0f * sqrtf(s11 * s22 - s12 * s21);
    if (isnan(Vb)) Vb = 0.0f;

    float KFIoU = Vb * rcp_nr(Vb_p + Vb_t - Vb + EPS_C);
    return fmaxf(xy_loss + (1.0f - KFIoU), 0.0f);
}

__global__ void kf_zero(float* __restrict__ ws) {
    if (threadIdx.x == 0 && blockIdx.x == 0) ws[0] = 0.0f;
}

__global__ void __launch_bounds__(256)
kf_main(const float* __restrict__ pred, const float* __restrict__ target,
        const float* __restrict__ ps, const float* __restrict__ ts,
        float* __restrict__ ws, int N) {
    int tid    = blockIdx.x * blockDim.x + threadIdx.x;
    int stride = gridDim.x * blockDim.x;
    int G      = (N + 3) >> 2;   // groups of 4 rows = 80 B = 5 aligned float4 per array

    float acc0 = 0.0f, acc1 = 0.0f;

    for (int g = tid; g < G; g += stride) {
        int r0 = g << 2;
        if (r0 + 4 <= N) {
            // ---- vector path: 5x global_load_b128 per array (16B-aligned) ----
            const v4f* vp = (const v4f*)(pred   + 20LL * g);
            const v4f* vt = (const v4f*)(target + 20LL * g);
            const v4f* vs = (const v4f*)(ps     + 20LL * g);
            const v4f* vu = (const v4f*)(ts     + 20LL * g);
            v4f qp[5], qt[5], qs[5], qu[5];
#pragma unroll
            for (int k = 0; k < 5; ++k) {
                qp[k] = vp[k]; qt[k] = vt[k]; qs[k] = vs[k]; qu[k] = vu[k];
            }
#pragma unroll
            for (int r = 0; r < 4; ++r) {
                float p[5], t[5], sp[5], st[5];
#pragma unroll
                for (int e = 0; e < 5; ++e) {
                    int idx = 5 * r + e;          // constant after unroll
                    p[e]  = qp[idx >> 2][idx & 3];
                    t[e]  = qt[idx >> 2][idx & 3];
                    sp[e] = qs[idx >> 2][idx & 3];
                    st[e] = qu[idx >> 2][idx & 3];
                }
                float l = loss_from(p, t, sp, st);
                if (r & 1) acc1 += l; else acc0 += l;
            }
        } else {
            // ---- tail: at most one group per grid, scalar loads ----
            for (int r = r0; r < N; ++r) {
                float p[5], t[5], sp[5], st[5];
                long long b = 5LL * r;
#pragma unroll
                for (int e = 0; e < 5; ++e) {
                    p[e]  = pred[b + e];
                    t[e]  = target[b + e];
                    sp[e] = ps[b + e];
                    st[e] = ts[b + e];
                }
                acc0 += loss_from(p, t, sp, st);
            }
        }
    }

    // --- Wave32 reduction via the matrix unit ---
    // A (16x4 f32, 2 VGPRs/lane) = the 64 partials of this wave.
    // B (4x16 f32, all ones)     -> D[m][n] = rowsum_m replicated over n.
    v2f a; a.x = acc0; a.y = acc1;
    v2f b; b.x = 1.0f; b.y = 1.0f;
    v8f c = {};
    c = __builtin_amdgcn_wmma_f32_16x16x4_f32(
            /*neg_a=*/false, a, /*neg_b=*/false, b,
            /*c_mod=*/(short)0, c, /*reuse_a=*/false, /*reuse_b=*/false);
    float s = c[0] + c[1] + c[2] + c[3] + c[4] + c[5] + c[6] + c[7];
    s += __shfl_xor(s, 16, 32);   // S_lo + S_hi = full wave sum in every lane

    // --- Cross-wave reduction in LDS (8 waves per 256-thread block) ---
    __shared__ float wsum[8];
    int lane = threadIdx.x & 31;
    int wave = threadIdx.x >> 5;
    if (lane == 0) wsum[wave] = s;
    __syncthreads();
    if (threadIdx.x == 0) {
        float bs = 0.0f;
        int nw = blockDim.x >> 5;
        for (int w = 0; w < nw; ++w) bs += wsum[w];
        atomicAdd(ws, bs);
    }
}

__global__ void kf_final(const float* __restrict__ ws, float* __restrict__ out, float invN) {
    if (threadIdx.x == 0 && blockIdx.x == 0) out[0] = ws[0] * invN;
}

extern "C" void kernel_launch(void* const* d_in, const int* in_sizes, int n_in,
                              void* d_out, int out_size, void* d_ws, size_t ws_size,
                              hipStream_t stream) {
    const float* pred   = (const float*)d_in[0];
    const float* target = (const float*)d_in[1];
    const float* ps     = (const float*)d_in[2];
    const float* ts     = (const float*)d_in[3];
    float* out = (float*)d_out;
    float* ws  = (float*)d_ws;

    int N = in_sizes[0] / 5;
    int G = (N + 3) / 4;

    kf_zero<<<1, 64, 0, stream>>>(ws);

    int block = 256;
    int grid  = (G + block - 1) / block;
    if (grid > 2048) grid = 2048;   // grid-stride: ~1 group (4 rows)/thread at N=2M
    if (grid < 1) grid = 1;
    kf_main<<<grid, block, 0, stream>>>(pred, target, ps, ts, ws, N);

    kf_final<<<1, 64, 0, stream>>>(ws, out, 1.0f / (float)N);
}